// SimpleGroupedPattern_25769803776599
// MI455X (gfx1250) — compile-verified
//
#include <hip/hip_runtime.h>

typedef __attribute__((ext_vector_type(2))) float v2f;
typedef __attribute__((ext_vector_type(8))) float v8f;
typedef __attribute__((ext_vector_type(4))) int   v4i;

#define AS1 __attribute__((address_space(1)))
#define AS3 __attribute__((address_space(3)))

#if defined(__HIP_DEVICE_COMPILE__)
#if !__has_builtin(__builtin_amdgcn_wmma_f32_16x16x4_f32)
#error "missing __builtin_amdgcn_wmma_f32_16x16x4_f32 for gfx1250"
#endif
#endif

// Async global->LDS path (CDNA5 GLOBAL_LOAD_ASYNC_TO_LDS_B128, ASYNCcnt).
// __has_builtin is evaluated per compile pass; host pass falls back harmlessly.
#if defined(__HIP_DEVICE_COMPILE__) && __has_builtin(__builtin_amdgcn_global_load_async_to_lds_b128)
#define USE_ASYNC_LDS 1
#else
#define USE_ASYNC_LDS 0
#endif

#if USE_ASYNC_LDS
#if __has_builtin(__builtin_amdgcn_s_wait_asynccnt)
#define WAIT_ASYNCCNT() __builtin_amdgcn_s_wait_asynccnt(0)
#else
#define WAIT_ASYNCCNT() asm volatile("s_wait_asynccnt 0x0" ::: "memory")
#endif
#endif

// Problem shape (from reference)
constexpr int G  = 8;
constexpr int K  = 4096;
constexpr int N  = 4096;
constexpr int MG = 2048;          // rows per group

// Tiling: block tile 128x256, 8 waves in 2x4 grid, wave tile 64x64 (4x4 WMMA tiles)
constexpr int BM = 128;
constexpr int BN = 256;
constexpr int BK = 16;
constexpr int LDA = BK + 4;       // 20 floats: conflict-free strided A-frag reads
constexpr int LDB = BN + 8;       // 264 floats: 2*264 % 64 == 16 -> half-waves on disjoint banks

__global__ __launch_bounds__(256)
void grouped_gemm_f32_wmma(const float* __restrict__ x,
                           const float* __restrict__ w,
                           float* __restrict__ out) {
  __shared__ float As[2][BM * LDA];   // A tile [BM][LDA]
  __shared__ float Bs[2][BK * LDB];   // B tile [BK][LDB]

  const int g  = blockIdx.z;
  const int m0 = blockIdx.y * BM;
  const int n0 = blockIdx.x * BN;

  const float* Ag = x + (size_t)(g * MG + m0) * K;
  const float* Bg = w + (size_t)g * K * N + n0;
  float*       Cg = out + (size_t)(g * MG + m0) * N + n0;

  const int tid  = threadIdx.x;
  const int lane = tid & 31;
  const int wid  = tid >> 5;
  const int wm   = (wid >> 2) * 64;   // wave row origin (2 wave-rows)
  const int wn   = (wid & 3) * 64;    // wave col origin (4 wave-cols)
  const int h    = lane >> 4;         // half-wave select
  const int l16  = lane & 15;

  v8f acc[4][4];
  #pragma unroll
  for (int i = 0; i < 4; ++i)
    #pragma unroll
    for (int j = 0; j < 4; ++j)
      acc[i][j] = (v8f){0.f, 0.f, 0.f, 0.f, 0.f, 0.f, 0.f, 0.f};

#if USE_ASYNC_LDS
  // Issue one tile's worth of async global->LDS copies (6 x b128 per thread).
  auto issue_tile = [&](int k0, int s) {
    #pragma unroll
    for (int i = 0; i < 2; ++i) {           // A: 128x16 f32 = 512 float4
      int idx = tid + i * 256;
      int r = idx >> 2, c = (idx & 3) << 2;
      __builtin_amdgcn_global_load_async_to_lds_b128(
          (AS1 v4i*)(uintptr_t)(Ag + (size_t)r * K + k0 + c),
          (AS3 v4i*)&As[s][r * LDA + c], 0, 0);
    }
    #pragma unroll
    for (int i = 0; i < 4; ++i) {           // B: 16x256 f32 = 1024 float4
      int idx = tid + i * 256;
      int r = idx >> 6, c = (idx & 63) << 2;
      __builtin_amdgcn_global_load_async_to_lds_b128(
          (AS1 v4i*)(uintptr_t)(Bg + (size_t)(k0 + r) * N + c),
          (AS3 v4i*)&Bs[s][r * LDB + c], 0, 0);
    }
  };
#else
  float4 ra[2], rb[4];
  auto load_tile = [&](int k0) {
    #pragma unroll
    for (int i = 0; i < 2; ++i) {
      int idx = tid + i * 256;
      int r = idx >> 2, c = (idx & 3) << 2;
      ra[i] = *(const float4*)(Ag + (size_t)r * K + k0 + c);
    }
    #pragma unroll
    for (int i = 0; i < 4; ++i) {
      int idx = tid + i * 256;
      int r = idx >> 6, c = (idx & 63) << 2;
      rb[i] = *(const float4*)(Bg + (size_t)(k0 + r) * N + c);
    }
  };
  auto store_tile = [&](int s) {
    #pragma unroll
    for (int i = 0; i < 2; ++i) {
      int idx = tid + i * 256;
      int r = idx >> 2, c = (idx & 3) << 2;
      *(float4*)&As[s][r * LDA + c] = ra[i];
    }
    #pragma unroll
    for (int i = 0; i < 4; ++i) {
      int idx = tid + i * 256;
      int r = idx >> 6, c = (idx & 63) << 2;
      *(float4*)&Bs[s][r * LDB + c] = rb[i];
    }
  };
#endif

  // 64 WMMAs per stage per wave (4 k-steps x 16 tiles)
  auto compute_stage = [&](int cur) {
    #pragma unroll
    for (int kk = 0; kk < BK; kk += 4) {
      v2f a[4], b[4];
      #pragma unroll
      for (int i = 0; i < 4; ++i)        // A VGPR v = A[m][kk + v + 2h]
        a[i] = *(const v2f*)&As[cur][(wm + i * 16 + l16) * LDA + kk + 2 * h];
      #pragma unroll
      for (int j = 0; j < 4; ++j) {      // B VGPR v = B[kk + v + 2h][n]
        int n = wn + j * 16 + l16;
        b[j].x = Bs[cur][(kk + 2 * h + 0) * LDB + n];
        b[j].y = Bs[cur][(kk + 2 * h + 1) * LDB + n];
      }
      #pragma unroll
      for (int i = 0; i < 4; ++i)
        #pragma unroll
        for (int j = 0; j < 4; ++j)
          acc[i][j] = __builtin_amdgcn_wmma_f32_16x16x4_f32(
              false, a[i], false, b[j], (short)0, acc[i][j], false, false);
    }
  };

  constexpr int NT = K / BK;              // 256 stages

#if USE_ASYNC_LDS
  issue_tile(0, 0);
  for (int kt = 0; kt < NT; ++kt) {
    const int cur = kt & 1;
    WAIT_ASYNCCNT();                      // my async writes into buf cur landed
    __syncthreads();                      // everyone's landed; reads of 1-cur done
    if (kt + 1 < NT) issue_tile((kt + 1) * BK, 1 - cur);
    compute_stage(cur);
  }
#else
  load_tile(0);
  store_tile(0);
  __syncthreads();
  for (int kt = 0; kt < NT; ++kt) {
    const int cur = kt & 1;
    if (kt + 1 < NT) load_tile((kt + 1) * BK);
    compute_stage(cur);
    if (kt + 1 < NT) store_tile(1 - cur);
    __syncthreads();
  }
#endif

  // Epilogue: C VGPR r -> row (r + 8h), col l16 within each 16x16 tile
  #pragma unroll
  for (int i = 0; i < 4; ++i)
    #pragma unroll
    for (int j = 0; j < 4; ++j)
      #pragma unroll
      for (int r = 0; r < 8; ++r) {
        int row = wm + i * 16 + 8 * h + r;
        int col = wn + j * 16 + l16;
        Cg[(size_t)row * N + col] = acc[i][j][r];
      }
}

extern "C" void kernel_launch(void* const* d_in, const int* in_sizes, int n_in,
                              void* d_out, int out_size, void* d_ws, size_t ws_size,
                              hipStream_t stream) {
  const float* x = (const float*)d_in[0];   // [16384, 4096] f32
  const float* w = (const float*)d_in[1];   // [8, 4096, 4096] f32
  float* out = (float*)d_out;               // [16384, 4096] f32

  dim3 grid(N / BN, MG / BM, G);            // (16, 16, 8)
  grouped_gemm_f32_wmma<<<grid, 256, 0, stream>>>(x, w, out);
}